// Quantizer_72121090834967
// MI455X (gfx1250) — compile-verified
//
#include <hip/hip_runtime.h>
#include <stdint.h>

#define COUT   1024
#define CIN    1024
#define KSZ    9
#define ROWLEN (CIN * KSZ)        /* 9216  */
#define NGROUP (COUT * CIN)       /* 1048576 */
#define NELEM  (COUT * ROWLEN)    /* 9437184 */

typedef float v8f  __attribute__((ext_vector_type(8)));
typedef float v2f  __attribute__((ext_vector_type(2)));
typedef uint32_t u32x4 __attribute__((ext_vector_type(4)));
typedef int      i32x8 __attribute__((ext_vector_type(8)));
typedef int      i32x4 __attribute__((ext_vector_type(4)));
typedef unsigned long long u64;

// ---------------------------------------------------------------------------
// Kernel 1: per-output-channel min/max -> scale, zero-point
// ---------------------------------------------------------------------------
__global__ __launch_bounds__(256) void k_rowstats(const float* __restrict__ w,
                                                  float* __restrict__ scale,
                                                  float* __restrict__ zp) {
  int row = blockIdx.x;
  const float* p = w + (size_t)row * ROWLEN;
  float mn = 1e30f, mx = -1e30f;
  for (int i = threadIdx.x; i < ROWLEN; i += 256) {
    float v = p[i];
    mn = fminf(mn, v);
    mx = fmaxf(mx, v);
  }
  __shared__ float smn[256], smx[256];
  smn[threadIdx.x] = mn; smx[threadIdx.x] = mx;
  __syncthreads();
  for (int s = 128; s > 0; s >>= 1) {
    if (threadIdx.x < s) {
      smn[threadIdx.x] = fminf(smn[threadIdx.x], smn[threadIdx.x + s]);
      smx[threadIdx.x] = fmaxf(smx[threadIdx.x], smx[threadIdx.x + s]);
    }
    __syncthreads();
  }
  if (threadIdx.x == 0) {
    float rng = fmaxf(smx[0] - smn[0], 1e-8f);
    float sc  = 15.0f / rng;
    scale[row] = sc;
    zp[row]    = rintf(sc * smn[0]) + 8.0f;
  }
}

// ---------------------------------------------------------------------------
// Kernel 2: SQuant-K pass. One thread per (Cout,Cin) group of 9 elements.
// Emits: post-flip r (int8), handed-off up/dn candidate values (int8, with
// li-revert applied), per-group residual error sum, and sparse boundary
// priorities (value + in-group position) for the up and dn arrays.
// ---------------------------------------------------------------------------
__global__ __launch_bounds__(256) void k_pass1(
    const float* __restrict__ w,
    const float* __restrict__ scale, const float* __restrict__ zp,
    int8_t* __restrict__ r8, int8_t* __restrict__ up8, int8_t* __restrict__ dn8,
    float* __restrict__ esum,
    float* __restrict__ upval, float* __restrict__ dnval,
    uint8_t* __restrict__ uppos, uint8_t* __restrict__ dnpos) {
  int g = blockIdx.x * 256 + threadIdx.x;
  if (g >= NGROUP) return;
  int co = g >> 10;
  float sc = scale[co], z = zp[co];
  const float* p = w + (size_t)g * KSZ;

  float r[KSZ], err[KSZ], upn[KSZ], dnn[KSZ], upe[KSZ], dne[KSZ];
  float pu[KSZ], pd[KSZ];
  float e = 0.f;
  for (int i = 0; i < KSZ; ++i) {
    float q  = sc * p[i] - z;
    float ri = rintf(q);            // round half to even, matches jnp.round
    float er = ri - q;
    bool cu = (er < 0.f) && (q < 7.0f);
    bool cd = (er > 0.f) && (q > -8.0f);
    r[i] = ri; err[i] = er;
    upn[i] = cu ? ri + 1.f : ri;
    upe[i] = cu ? er + 1.f : 0.f;
    pu[i]  = cu ? -er      : 0.f;
    dnn[i] = cd ? ri - 1.f : ri;
    dne[i] = cd ? er - 1.f : 0.f;
    pd[i]  = cd ? er       : 0.f;
    e += er;
  }
  int  nflip = (int)rintf(fabsf(e));
  bool is_up = (e < 0.f);

  float pri[KSZ];
  for (int i = 0; i < KSZ; ++i) pri[i] = is_up ? pu[i] : pd[i];

  // stable insertion sort: descending priority, ties by ascending index
  int ord[KSZ];
  for (int i = 0; i < KSZ; ++i) ord[i] = i;
  for (int i = 1; i < KSZ; ++i) {
    int o = ord[i]; float pv = pri[o]; int j = i - 1;
    while (j >= 0 && pri[ord[j]] < pv) { ord[j + 1] = ord[j]; --j; }
    ord[j + 1] = o;
  }

  float nr[KSZ], ne[KSZ];
  for (int i = 0; i < KSZ; ++i) { nr[i] = r[i]; ne[i] = err[i]; }
  for (int t = 0; t < nflip && t < KSZ; ++t) {
    int i = ord[t];
    nr[i] = is_up ? upn[i] : dnn[i];
    ne[i] = is_up ? upe[i] : dne[i];
  }

  int ni = ord[nflip < KSZ ? nflip : KSZ - 1];
  int lt = nflip - 1; if (lt < 0) lt = 0; if (lt > KSZ - 1) lt = KSZ - 1;
  int li = ord[lt];
  bool has_last = (nflip > 0);
  bool has_next = (nflip < KSZ);
  float next_pri = has_next ? pri[ni] : 0.f;
  float last_pri = has_last ? 1.f - pri[li] : 0.f;

  // revert candidates for next pass (restores pre-flip value at li)
  if (is_up && has_last)  dnn[li] = r[li];
  if (!is_up && has_last) upn[li] = r[li];

  float es = 0.f;
  for (int i = 0; i < KSZ; ++i) es += ne[i];
  esum[g] = es;

  size_t base = (size_t)g * KSZ;
  for (int i = 0; i < KSZ; ++i) {
    r8[base + i]  = (int8_t)(int)nr[i];
    up8[base + i] = (int8_t)(int)upn[i];
    dn8[base + i] = (int8_t)(int)dnn[i];
  }
  if (is_up) { upval[g] = next_pri; uppos[g] = (uint8_t)ni;
               dnval[g] = last_pri; dnpos[g] = (uint8_t)li; }
  else       { upval[g] = last_pri; uppos[g] = (uint8_t)li;
               dnval[g] = next_pri; dnpos[g] = (uint8_t)ni; }
}

// ---------------------------------------------------------------------------
// Kernel 3: exact f32 row sums of the 1024x1024 esum matrix via WMMA.
// D = A(16x4 esums) * B(4x16 ones) + C ; every column of D is the row sum.
// One wave per 16 rows. EXEC is all-ones through the WMMA loop.
// ---------------------------------------------------------------------------
__global__ __launch_bounds__(32) void k_rowsum_wmma(const float* __restrict__ esum,
                                                    float* __restrict__ e_out) {
  int m0   = blockIdx.x * 16;
  int lane = threadIdx.x;                 // 0..31
  int row  = m0 + (lane & 15);
  int koff = (lane >> 4) * 2;             // lanes 0-15: K0,K1 ; lanes 16-31: K2,K3
  const float* base = esum + (size_t)row * CIN;

  v8f acc = {};
  v2f b; b[0] = 1.0f; b[1] = 1.0f;        // all-ones B: layout-independent
  for (int k = 0; k < CIN; k += 4) {
    v2f a;
    a[0] = base[k + koff];
    a[1] = base[k + koff + 1];
    acc = __builtin_amdgcn_wmma_f32_16x16x4_f32(
        /*neg_a=*/false, a, /*neg_b=*/false, b,
        /*c_mod=*/(short)0, acc, /*reuse_a=*/false, /*reuse_b=*/false);
  }
  // C/D layout: VGPR i holds D[i][lane] (lanes 0-15) and D[i+8][lane-16]
  if (lane == 0)       { for (int i = 0; i < 8; ++i) e_out[m0 + i]     = acc[i]; }
  else if (lane == 16) { for (int i = 0; i < 8; ++i) e_out[m0 + 8 + i] = acc[i]; }
}

// ---------------------------------------------------------------------------
// Kernel 4: SQuant-C pass + dequantize. One block per output channel.
// The row's r bytes are DMA'd into LDS by the Tensor Data Mover while the
// block compacts + bitonic-sorts the sparse priorities; flips patch LDS.
// ---------------------------------------------------------------------------
__global__ __launch_bounds__(256) void k_pass2(
    const float* __restrict__ e_arr,
    const float* __restrict__ upval, const float* __restrict__ dnval,
    const uint8_t* __restrict__ uppos, const uint8_t* __restrict__ dnpos,
    const int8_t* __restrict__ r8, const int8_t* __restrict__ up8,
    const int8_t* __restrict__ dn8,
    const float* __restrict__ scale, const float* __restrict__ zp,
    float* __restrict__ out) {
  int row = blockIdx.x;
  int t   = threadIdx.x;
  size_t rbase = (size_t)row * ROWLEN;

  __shared__ u64      keys[1024];
  __shared__ uint32_t posmask[288];     // 9216-bit bitmap of positive-pri positions
  __shared__ int      pref[289];
  __shared__ int      cnt;
  __shared__ int8_t   smem_r[ROWLEN];

  // --- TDM: async-load the 9216-byte r row into LDS (wave 0 only issues) ---
  if ((t >> 5) == 0) {
    uint64_t ga  = (uint64_t)(uintptr_t)(r8 + rbase);
    uint32_t lds = (uint32_t)(uintptr_t)(void*)smem_r;
    u32x4 g0;
    g0[0] = 1u;                                         // count=1, user D#
    g0[1] = lds;                                        // lds_addr
    g0[2] = (uint32_t)ga;                               // global_addr[31:0]
    g0[3] = ((uint32_t)(ga >> 32) & 0x01FFFFFFu) | (2u << 30); // addr[56:32], type=2
    i32x8 g1;
    g1[0] = (int)(2u << 16);                            // data_size = 4B dwords
    g1[1] = (int)((2304u & 0xFFFFu) << 16);             // tensor_dim0[15:0]
    g1[2] = (int)((1u << 16) | (2304u >> 16));          // tensor_dim1=1 | dim0[31:16]
    g1[3] = (int)(2304u << 16);                         // tile_dim0 = 2304 dwords
    g1[4] = 1;                                          // tile_dim1 = 1
    g1[5] = 2304;                                       // tensor_dim0_stride
    g1[6] = 0;
    g1[7] = 0;
    i32x4 gz4; gz4[0] = 0; gz4[1] = 0; gz4[2] = 0; gz4[3] = 0;
    i32x8 gz8;
    for (int i = 0; i < 8; ++i) gz8[i] = 0;
    __builtin_amdgcn_tensor_load_to_lds(g0, g1, gz4, gz4, gz8, 0);
  }

  if (t == 0) cnt = 0;
  for (int i = t; i < 288; i += 256) posmask[i] = 0u;
  __syncthreads();

  float e     = e_arr[row];
  int   nflip = (int)rintf(fabsf(e));
  if (nflip > ROWLEN) nflip = ROWLEN;
  bool  is_up = (e < 0.f);
  int   gbase = row * CIN;

  // compact positive priorities; key = (pri_bits << 32) | ~pos  (pri desc, pos asc)
  for (int g = t; g < CIN; g += 256) {
    float v = is_up ? upval[gbase + g] : dnval[gbase + g];
    int   p = is_up ? (int)uppos[gbase + g] : (int)dnpos[gbase + g];
    if (v > 0.f) {
      int slot = atomicAdd(&cnt, 1);
      uint32_t posrow = (uint32_t)(g * KSZ + p);
      keys[slot] = ((u64)__float_as_uint(v) << 32) | (u64)(uint32_t)(~posrow);
      atomicOr(&posmask[posrow >> 5], 1u << (posrow & 31));
    }
  }
  __syncthreads();
  int P = cnt;
  for (int i = P + t; i < 1024; i += 256) keys[i] = 0ull;  // pad sorts last
  __syncthreads();

  // bitonic sort, descending
  for (int ksz = 2; ksz <= 1024; ksz <<= 1) {
    for (int j = ksz >> 1; j > 0; j >>= 1) {
      for (int i = t; i < 1024; i += 256) {
        int ixj = i ^ j;
        if (ixj > i) {
          u64 a = keys[i], b = keys[ixj];
          bool desc = ((i & ksz) == 0);
          if (desc ? (a < b) : (a > b)) { keys[i] = b; keys[ixj] = a; }
        }
      }
      __syncthreads();
    }
  }

  // prefix of positive-position popcounts (for zero-priority fill ordering)
  if (t == 0) {
    int s = 0;
    for (int w2 = 0; w2 < 288; ++w2) { pref[w2] = s; s += __popc(posmask[w2]); }
    pref[288] = s;
  }
  __syncthreads();

  // wait for the TDM row, then patch flipped positions in LDS
  __builtin_amdgcn_s_wait_tensorcnt(0);
  __syncthreads();

  int npos = (nflip < P) ? nflip : P;
  for (int i = t; i < npos; i += 256) {
    uint32_t posrow = ~(uint32_t)(keys[i] & 0xFFFFFFFFull);
    smem_r[posrow] = is_up ? up8[rbase + posrow] : dn8[rbase + posrow];
  }
  if (nflip > P) {
    int rem = nflip - P;
    for (int j2 = t; j2 < ROWLEN; j2 += 256) {
      uint32_t m = posmask[j2 >> 5];
      if (!((m >> (j2 & 31)) & 1u)) {
        int zr = j2 - (pref[j2 >> 5] + __popc(m & ((1u << (j2 & 31)) - 1u)));
        if (zr < rem)
          smem_r[j2] = is_up ? up8[rbase + j2] : dn8[rbase + j2];
      }
    }
  }
  __syncthreads();

  // clip + dequantize + stream out
  float sc = scale[row], z = zp[row];
  for (int j2 = t; j2 < ROWLEN; j2 += 256) {
    int rv = (int)smem_r[j2];
    rv = rv < -8 ? -8 : (rv > 7 ? 7 : rv);
    out[rbase + j2] = ((float)rv + z) / sc;
  }
}

// ---------------------------------------------------------------------------
extern "C" void kernel_launch(void* const* d_in, const int* in_sizes, int n_in,
                              void* d_out, int out_size, void* d_ws, size_t ws_size,
                              hipStream_t stream) {
  const float* w = (const float*)d_in[0];
  float* out = (float*)d_out;
  uint8_t* ws = (uint8_t*)d_ws;

  size_t o = 0;
  float*   scale = (float*)(ws + o);   o += 4096;
  float*   zp    = (float*)(ws + o);   o += 4096;
  float*   e_arr = (float*)(ws + o);   o += 4096;
  o = (o + 255) & ~(size_t)255;
  float*   esum  = (float*)(ws + o);   o += (size_t)NGROUP * 4;
  float*   upval = (float*)(ws + o);   o += (size_t)NGROUP * 4;
  float*   dnval = (float*)(ws + o);   o += (size_t)NGROUP * 4;
  uint8_t* uppos = (uint8_t*)(ws + o); o += (size_t)NGROUP;
  uint8_t* dnpos = (uint8_t*)(ws + o); o += (size_t)NGROUP;
  o = (o + 255) & ~(size_t)255;
  int8_t*  r8    = (int8_t*)(ws + o);  o += (size_t)NELEM;
  o = (o + 255) & ~(size_t)255;
  int8_t*  up8   = (int8_t*)(ws + o);  o += (size_t)NELEM;
  o = (o + 255) & ~(size_t)255;
  int8_t*  dn8   = (int8_t*)(ws + o);  o += (size_t)NELEM;
  (void)ws_size; (void)in_sizes; (void)n_in; (void)out_size;

  k_rowstats<<<COUT, 256, 0, stream>>>(w, scale, zp);
  k_pass1<<<NGROUP / 256, 256, 0, stream>>>(w, scale, zp, r8, up8, dn8,
                                            esum, upval, dnval, uppos, dnpos);
  k_rowsum_wmma<<<COUT / 16, 32, 0, stream>>>(esum, e_arr);
  k_pass2<<<COUT, 256, 0, stream>>>(e_arr, upval, dnval, uppos, dnpos,
                                    r8, up8, dn8, scale, zp, out);
}